// GNNModel_38620345925753
// MI455X (gfx1250) — compile-verified
//
#include <hip/hip_runtime.h>
#include <stdint.h>

#define BATCH 8
#define NANCH 32768
#define CH 105
#define TOPK 1024
#define CHUNK 4096
#define NCHUNK (NANCH / CHUNK)     // 8
#define CAND (NCHUNK * TOPK)       // 8192
#define CONF_THRE 0.001f
#define NMS_THRE 0.65f
#define CLS_OFF 241.0f             // MAX_DIM + 1

#if defined(__HIP_DEVICE_COMPILE__) && __has_builtin(__builtin_amdgcn_global_load_async_to_lds_b128)
#define HAVE_ASYNC 1
#else
#define HAVE_ASYNC 0
#endif

// ---- monotone float <-> uint mapping (unsigned order == float order) ----
__device__ __forceinline__ uint32_t fmono(float f) {
  uint32_t b = __float_as_uint(f);
  return (b & 0x80000000u) ? ~b : (b | 0x80000000u);
}
__device__ __forceinline__ float fmono_inv(uint32_t m) {
  uint32_t b = (m & 0x80000000u) ? (m & 0x7FFFFFFFu) : ~m;
  return __uint_as_float(b);
}
__device__ __forceinline__ uint32_t ballot32(bool p) {
#if defined(__HIP_DEVICE_COMPILE__) && __has_builtin(__builtin_amdgcn_ballot_w32)
  return __builtin_amdgcn_ballot_w32(p);
#else
  return (uint32_t)__ballot(p);
#endif
}
__device__ __forceinline__ uint64_t shflxor_u64(uint64_t v, int m) {
  uint32_t lo = __shfl_xor((uint32_t)v, m, 32);
  uint32_t hi = __shfl_xor((uint32_t)(v >> 32), m, 32);
  return ((uint64_t)hi << 32) | lo;
}

#if HAVE_ASYNC
typedef int v4i_t __attribute__((ext_vector_type(4)));
typedef __attribute__((address_space(1))) v4i_t* g4ptr_t;   // global int4*
typedef __attribute__((address_space(3))) v4i_t* l4ptr_t;   // LDS int4*
__device__ __forceinline__ void async_b128(const void* g, void* l) {
  __builtin_amdgcn_global_load_async_to_lds_b128((g4ptr_t)g, (l4ptr_t)l, 0, 0);
}
__device__ __forceinline__ void wait_async0() {
#if __has_builtin(__builtin_amdgcn_s_wait_asynccnt)
  __builtin_amdgcn_s_wait_asynccnt(0);
#else
  asm volatile("s_wait_asynccnt 0" ::: "memory");
#endif
}
#endif

// ---------------------------------------------------------------------------
// Kernel 1: per-anchor class max/argmax + sortable key. One wave per anchor.
// key = (mono(maskedScore) << 32) | ~idx  (descending key order == top_k order)
// keysInv stores ~key so an ASCENDING sort yields descending scores.
// ---------------------------------------------------------------------------
__global__ __launch_bounds__(256) void k_score(const float* __restrict__ pred,
                                               uint64_t* __restrict__ keysInv,
                                               int* __restrict__ labels) {
  const int lane = threadIdx.x & 31;
  const int a = blockIdx.x * 8 + (threadIdx.x >> 5);   // anchor id in [0, B*N)
  const float* p = pred + (size_t)a * CH;
  __builtin_prefetch(p + 8 * CH, 0, 1);                // global_prefetch_b8
  float v0 = p[lane];
  float v1 = p[lane + 32];
  float v2 = p[lane + 64];
  float v3 = (lane < CH - 96) ? p[lane + 96] : 0.0f;   // channels 96..104

  // pack (score, ~class): max picks highest score, then smallest class (ties)
  uint64_t best = 0;
  if (lane >= 5)
    best = ((uint64_t)fmono(v0) << 32) | (uint32_t)~(uint32_t)(lane - 5);
  uint64_t c = ((uint64_t)fmono(v1) << 32) | (uint32_t)~(uint32_t)(lane + 27);
  if (c > best) best = c;
  c = ((uint64_t)fmono(v2) << 32) | (uint32_t)~(uint32_t)(lane + 59);
  if (c > best) best = c;
  if (lane < CH - 96) {
    c = ((uint64_t)fmono(v3) << 32) | (uint32_t)~(uint32_t)(lane + 91);
    if (c > best) best = c;
  }
  for (int m = 16; m; m >>= 1) {
    uint64_t o = shflxor_u64(best, m);
    if (o > best) best = o;
  }
  float obj = __shfl(v0, 4, 32);   // channel 4 = objectness
  if (lane == 0) {
    float conf = fmono_inv((uint32_t)(best >> 32));
    int cls = (int)~(uint32_t)best;
    float score = obj * conf;
    float masked = (score * conf >= CONF_THRE) ? score : -1.0f;
    uint32_t n = (uint32_t)(a & (NANCH - 1));
    uint64_t key = ((uint64_t)fmono(masked) << 32) | (uint32_t)~n;
    keysInv[a] = ~key;
    labels[a] = cls;
  }
}

// ---------------------------------------------------------------------------
// Bitonic sort (ascending) of M u64 keys in LDS, blockDim-strided.
// ---------------------------------------------------------------------------
__device__ void bitonic_asc(uint64_t* s, int M) {
  for (int k = 2; k <= M; k <<= 1) {
    for (int j = k >> 1; j > 0; j >>= 1) {
      for (int i = threadIdx.x; i < M; i += blockDim.x) {
        int ixj = i ^ j;
        if (ixj > i) {
          uint64_t a = s[i], b = s[ixj];
          if (((i & k) == 0) == (a > b)) { s[i] = b; s[ixj] = a; }
        }
      }
      __syncthreads();
    }
  }
}

// Kernel 2a: sort each 4096-chunk ascending, keep its best (smallest keysInv)
// 1024 as a SORTED run of candidates.
__global__ __launch_bounds__(1024) void k_sort1(const uint64_t* __restrict__ keysInv,
                                                uint64_t* __restrict__ cand) {
  __shared__ uint64_t s[CHUNK];                        // 32 KB LDS
  const int b = blockIdx.x / NCHUNK, cidx = blockIdx.x % NCHUNK;
  const uint64_t* src = keysInv + (size_t)b * NANCH + (size_t)cidx * CHUNK;
#if HAVE_ASYNC
  for (int i = threadIdx.x * 2; i < CHUNK; i += 2048) async_b128(src + i, s + i);
  wait_async0();
#else
  for (int i = threadIdx.x; i < CHUNK; i += 1024) s[i] = src[i];
#endif
  __syncthreads();
  bitonic_asc(s, CHUNK);
  cand[(size_t)b * CAND + cidx * TOPK + threadIdx.x] = s[threadIdx.x];
}

// Kernel 2b: fold the 8 sorted 1024-runs into the exact global top-1024.
// For two ascending runs a,b of length n: min(a[i], b[n-1-i]) is the n
// smallest of the union AND bitonic -> one 10-phase half-cleaner re-sorts it.
// 7 folds x (min + 10 phases) on 1024 elems (1/thread) vs a 91-phase 8192 sort.
__global__ __launch_bounds__(1024) void k_sort2(const uint64_t* __restrict__ cand,
                                                int* __restrict__ topIdx,
                                                float* __restrict__ topScore) {
  __shared__ uint64_t res[TOPK];                       // 8 KB
  __shared__ uint64_t run[TOPK];                       // 8 KB
  const int b = blockIdx.x;
  const int t = threadIdx.x;
  const uint64_t* src = cand + (size_t)b * CAND;
#if HAVE_ASYNC
  if (t < 512) async_b128(src + 2 * t, res + 2 * t);
#else
  res[t] = src[t];
#endif
  for (int r = 1; r < NCHUNK; ++r) {
#if HAVE_ASYNC
    if (t < 512) async_b128(src + (size_t)r * TOPK + 2 * t, run + 2 * t);
    wait_async0();
#else
    run[t] = src[(size_t)r * TOPK + t];
#endif
    __syncthreads();
    // keep the 1024 smallest of res ∪ run (forms a bitonic sequence)
    uint64_t a = res[t], o = run[TOPK - 1 - t];
    res[t] = (a < o) ? a : o;                          // only thread t touches res[t]
    __syncthreads();
    for (int j = TOPK >> 1; j > 0; j >>= 1) {          // bitonic half-cleaner
      if ((t & j) == 0) {
        uint64_t x = res[t], y = res[t + j];
        if (x > y) { res[t] = y; res[t + j] = x; }
      }
      __syncthreads();
    }
  }
  uint64_t key = ~res[t];                              // rank t (descending score)
  topScore[b * TOPK + t] = fmono_inv((uint32_t)(key >> 32));
  topIdx[b * TOPK + t] = (int)~(uint32_t)key;
}

// Kernel 3: gather kept anchors, build raw boxes + class-offset boxes + areas.
__global__ __launch_bounds__(256) void k_gather(const float* __restrict__ pred,
                                                const int* __restrict__ labels,
                                                const int* __restrict__ topIdx,
                                                float4* __restrict__ rawBox,
                                                float4* __restrict__ offBox,
                                                float* __restrict__ area,
                                                float* __restrict__ labK) {
  const int t = blockIdx.x * 256 + threadIdx.x;        // [0, B*TOPK)
  const int b = t >> 10;
  const int idx = topIdx[t];
  const float* p = pred + ((size_t)b * NANCH + (size_t)idx) * CH;
  float cx = p[0], cy = p[1], wd = p[2], ht = p[3];
  float x1 = cx - wd * 0.5f, y1 = cy - ht * 0.5f;
  float x2 = wd + x1, y2 = ht + y1;                    // matches reference: xy2 = wh + xy1
  int lab = labels[(size_t)b * NANCH + idx];
  float off = (float)lab * CLS_OFF;
  rawBox[t] = make_float4(x1, y1, x2, y2);
  float ox1 = x1 + off, oy1 = y1 + off, ox2 = x2 + off, oy2 = y2 + off;
  offBox[t] = make_float4(ox1, oy1, ox2, oy2);
  area[t] = (ox2 - ox1) * (oy2 - oy1);                 // area of offset boxes (as ref)
  labK[t] = (float)lab;
}

// Kernel 4: adjacency bitmask  mask[b][i][j/32] bit j = (IoU(i,j) > thre).
// Divide-free: inter/den > t  <=>  inter > t*den  (den > 0). Ballot packs 32/op.
__global__ __launch_bounds__(256) void k_iou(const float4* __restrict__ offBox,
                                             const float* __restrict__ area,
                                             uint32_t* __restrict__ mask) {
  const int t = blockIdx.x * 256 + threadIdx.x;
  const int j = t & (TOPK - 1);
  const int i = (t >> 10) & (TOPK - 1);
  const int b = t >> 20;
  const float4* ob = offBox + (size_t)b * TOPK;
  float4 bi = ob[i];                                   // uniform across wave -> broadcast
  float4 bj = ob[j];                                   // consecutive -> coalesced
  float ai = area[b * TOPK + i];
  float aj = area[b * TOPK + j];
  float ix1 = fmaxf(bi.x, bj.x), iy1 = fmaxf(bi.y, bj.y);
  float ix2 = fminf(bi.z, bj.z), iy2 = fminf(bi.w, bj.w);
  float iw = fmaxf(ix2 - ix1, 0.0f), ih = fmaxf(iy2 - iy1, 0.0f);
  float inter = iw * ih;
  bool cnd = inter > NMS_THRE * (ai + aj - inter + 1e-12f);
  uint32_t m = ballot32(cnd);
  if ((threadIdx.x & 31) == 0)
    mask[(size_t)b * (TOPK * 32) + (size_t)i * 32 + (j >> 5)] = m;
}

// Kernel 5: greedy NMS on the LDS-resident 128 KB bitmask (one wave32/batch),
// then masked writeback of boxes/score/label. supp/keep/valid are bit-words
// distributed one per lane; pivot bits broadcast via __shfl.
__global__ __launch_bounds__(32) void k_nms(const uint32_t* __restrict__ mask,
                                            const float* __restrict__ topScore,
                                            const float4* __restrict__ rawBox,
                                            const float* __restrict__ labK,
                                            float* __restrict__ out) {
  __shared__ uint32_t sm[TOPK * 32];                   // 128 KB LDS bitmask
  const int b = blockIdx.x;
  const int lane = threadIdx.x;
  const uint32_t* gm = mask + (size_t)b * (TOPK * 32);
#if HAVE_ASYNC
  for (int i = lane * 4; i < TOPK * 32; i += 128) async_b128(gm + i, sm + i);
  wait_async0();
#else
  for (int i = lane * 4; i < TOPK * 32; i += 128) {
    uint4 v = *(const uint4*)(gm + i);
    *(uint4*)(sm + i) = v;
  }
#endif
  __syncthreads();

  uint32_t validW = 0;                                 // lane L owns k in [32L, 32L+32)
  for (int t = 0; t < 32; ++t) {
    bool cnd = topScore[b * TOPK + t * 32 + lane] >= 0.0f;
    uint32_t w = ballot32(cnd);
    if (lane == t) validW = w;
  }
  uint32_t supp = 0, keep = 0;
  for (int i = 0; i < TOPK; ++i) {
    const int wl = i >> 5, bt = i & 31;
    uint32_t sw = __shfl(supp, wl, 32);
    uint32_t vw = __shfl(validW, wl, 32);
    if (((vw >> bt) & 1u) && !((sw >> bt) & 1u)) {     // uniform across wave
      supp |= sm[i * 32 + lane];                       // OR row i (incl. diagonal, as ref)
      if (lane == wl) keep |= (1u << bt);
    }
  }
  for (int t = 0; t < 32; ++t) {                       // coalesced writeback
    const int k = t * 32 + lane;
    uint32_t kw = __shfl(keep, t, 32);
    float kf = ((kw >> lane) & 1u) ? 1.0f : 0.0f;
    float4 bx = rawBox[b * TOPK + k];
    float sc = topScore[b * TOPK + k];
    float lb = labK[b * TOPK + k];
    float* o = out + ((size_t)b * TOPK + k) * 6;
    o[0] = bx.x * kf; o[1] = bx.y * kf; o[2] = bx.z * kf; o[3] = bx.w * kf;
    o[4] = sc * kf;   o[5] = lb * kf;
  }
}

extern "C" void kernel_launch(void* const* d_in, const int* in_sizes, int n_in,
                              void* d_out, int out_size, void* d_ws, size_t ws_size,
                              hipStream_t stream) {
  const float* pred = (const float*)d_in[0];
  float* out = (float*)d_out;
  uint8_t* w = (uint8_t*)d_ws;
  // workspace layout (~5.1 MB total)
  uint64_t* keysInv = (uint64_t*)(w + 0x000000);   // 2 MB
  int*      labels  = (int*)     (w + 0x200000);   // 1 MB
  uint64_t* cand    = (uint64_t*)(w + 0x300000);   // 512 KB
  int*      topIdx  = (int*)     (w + 0x380000);   // 32 KB
  float*    topScore= (float*)   (w + 0x388000);   // 32 KB
  float4*   rawBox  = (float4*)  (w + 0x390000);   // 128 KB
  float4*   offBox  = (float4*)  (w + 0x3B0000);   // 128 KB
  float*    area    = (float*)   (w + 0x3D0000);   // 32 KB
  float*    labK    = (float*)   (w + 0x3D8000);   // 32 KB
  uint32_t* mask    = (uint32_t*)(w + 0x3E0000);   // 1 MB

  k_score <<<(BATCH * NANCH) / 8, 256, 0, stream>>>(pred, keysInv, labels);
  k_sort1 <<<BATCH * NCHUNK, 1024, 0, stream>>>(keysInv, cand);
  k_sort2 <<<BATCH, 1024, 0, stream>>>(cand, topIdx, topScore);
  k_gather<<<(BATCH * TOPK) / 256, 256, 0, stream>>>(pred, labels, topIdx,
                                                     rawBox, offBox, area, labK);
  k_iou   <<<(BATCH * TOPK * TOPK) / 256, 256, 0, stream>>>(offBox, area, mask);
  k_nms   <<<BATCH, 32, 0, stream>>>(mask, topScore, rawBox, labK, out);
  (void)in_sizes; (void)n_in; (void)out_size; (void)ws_size;
}